// Detail_loss_50053548867701
// MI455X (gfx1250) — compile-verified
//
#include <hip/hip_runtime.h>
#include <hip/hip_bf16.h>

#define BATCH 32
#define HDIM  512
#define WDIM  512
#define NPIX  (HDIM * WDIM)          // 262144
#define NBINS 256
#define NT    254

typedef float v2f __attribute__((ext_vector_type(2)));
typedef float v8f __attribute__((ext_vector_type(8)));

// ---- CDNA5 async global->LDS path (guarded; falls back to plain loads) ----
#if defined(__has_builtin)
#  if __has_builtin(__builtin_amdgcn_global_load_async_to_lds_b32)
#    define HAVE_ASYNC_LDS 1
#  endif
#endif
#ifndef HAVE_ASYNC_LDS
#  define HAVE_ASYNC_LDS 0
#endif

#if HAVE_ASYNC_LDS
typedef __attribute__((address_space(1))) int gas_int;   // global (__device__) int*
typedef __attribute__((address_space(3))) int las_int;   // LDS (__shared__) int*
// generic global address bits == AS(1) bits; generic LDS address low 32 bits == LDS offset
#define TO_GLOBAL(p) ((gas_int*)(unsigned long long)(p))
#define TO_LDS(p)    ((las_int*)(unsigned int)(unsigned long long)(p))
__device__ __forceinline__ void wait_async0() {
#  if __has_builtin(__builtin_amdgcn_s_wait_asynccnt)
    __builtin_amdgcn_s_wait_asynccnt(0);
#  else
    asm volatile("s_wait_asynccnt 0" ::: "memory");
#  endif
}
#endif

// ---------------------------------------------------------------------------
// 0) zero the accumulator regions of the workspace (it is poisoned to 0xAA)
// ---------------------------------------------------------------------------
__global__ void init_kernel(float* __restrict__ hist, float* __restrict__ acc) {
    int i = blockIdx.x * blockDim.x + threadIdx.x;
    if (i < BATCH * NBINS) hist[i] = 0.0f;
    if (i < BATCH * 2)     acc[i]  = 0.0f;
}

// ---------------------------------------------------------------------------
// 1) 5x5 binary dilation, LDS-tiled. Each block stages a 68x36 halo tile of
//    labels into LDS (async DMA path when available), then computes a 64x32
//    output tile with a column-max factorization, storing uchar4 masks.
// ---------------------------------------------------------------------------
#define TW   64
#define TH   32
#define HALO 2
#define LW   (TW + 2 * HALO)   // 68
#define LH   (TH + 2 * HALO)   // 36

__global__ void dilate_kernel(const float* __restrict__ labels,
                              unsigned char* __restrict__ mask) {
    __shared__ float tile[LH][LW];          // 9792 B of the 320 KB WGP LDS
    const int tid = threadIdx.x;
    const int x0  = blockIdx.x * TW;
    const int y0  = blockIdx.y * TH;
    const int b   = blockIdx.z;
    const float* L = labels + (size_t)b * NPIX;

    // ---- stage tile + halo (zero padding outside the image) ----
    for (int e = tid; e < LW * LH; e += 256) {
        const int ly = e / LW;
        const int lx = e - ly * LW;
        const int gy = y0 + ly - HALO;
        const int gx = x0 + lx - HALO;
        const bool inb = (gy >= 0) & (gy < HDIM) & (gx >= 0) & (gx < WDIM);
#if HAVE_ASYNC_LDS
        if (inb) {
            __builtin_amdgcn_global_load_async_to_lds_b32(
                TO_GLOBAL(L + gy * WDIM + gx), TO_LDS(&tile[ly][lx]), 0, 0);
        } else {
            tile[ly][lx] = 0.0f;
        }
#else
        tile[ly][lx] = inb ? L[gy * WDIM + gx] : 0.0f;
#endif
    }
#if HAVE_ASYNC_LDS
    wait_async0();
#endif
    __syncthreads();

    // ---- 4 adjacent outputs per iteration: 8 column maxes -> 4 window maxes
    unsigned char* mrow = mask + (size_t)b * NPIX;
    for (int e = tid; e < (TW / 4) * TH; e += 256) {
        const int ty = e >> 4;              // e / 16
        const int tx = (e & 15) * 4;
        float cmax[8];
        #pragma unroll
        for (int c = 0; c < 8; ++c) {
            float mm = tile[ty + 0][tx + c];
            #pragma unroll
            for (int r = 1; r < 5; ++r) mm = fmaxf(mm, tile[ty + r][tx + c]);
            cmax[c] = mm;
        }
        uchar4 o;
        {
            float m0 = fmaxf(fmaxf(fmaxf(cmax[0], cmax[1]), fmaxf(cmax[2], cmax[3])), cmax[4]);
            float m1 = fmaxf(fmaxf(fmaxf(cmax[1], cmax[2]), fmaxf(cmax[3], cmax[4])), cmax[5]);
            float m2 = fmaxf(fmaxf(fmaxf(cmax[2], cmax[3]), fmaxf(cmax[4], cmax[5])), cmax[6]);
            float m3 = fmaxf(fmaxf(fmaxf(cmax[3], cmax[4]), fmaxf(cmax[5], cmax[6])), cmax[7]);
            o.x = m0 > 0.0f; o.y = m1 > 0.0f; o.z = m2 > 0.0f; o.w = m3 > 0.0f;
        }
        *(uchar4*)&mrow[(y0 + ty) * WDIM + x0 + tx] = o;
    }
}

// ---------------------------------------------------------------------------
// 2) per-batch 256-bin histogram (LDS privatized, float4/uchar4 streaming)
//    idx = clip(floor(img*mask*255 * 256/255), 0, 255) == min(255, floor(img*256))
// ---------------------------------------------------------------------------
#define HCHUNKS 16
#define HPP     (NPIX / HCHUNKS)     // 16384 pixels per block

__global__ void hist_kernel(const float* __restrict__ images,
                            const unsigned char* __restrict__ mask,
                            float* __restrict__ hist) {
    __shared__ float lh[NBINS];
    const int b   = blockIdx.y;
    const int tid = threadIdx.x;
    lh[tid] = 0.0f;
    __syncthreads();
    const int base = blockIdx.x * HPP;
    const float4* img4 = (const float4*)(images + (size_t)b * NPIX + base);
    const uchar4* mk4  = (const uchar4*)(mask   + (size_t)b * NPIX + base);
    for (int t = tid; t < HPP / 4; t += 256) {
        __builtin_prefetch(&img4[t + 256], 0, 0);
        const uchar4 m = mk4[t];
        const float4 v = img4[t];
        if (m.x | m.y | m.z | m.w) {
            if (m.x) { int i0 = (int)(v.x * 256.0f); i0 = i0 > 255 ? 255 : (i0 < 0 ? 0 : i0); atomicAdd(&lh[i0], 1.0f); }
            if (m.y) { int i1 = (int)(v.y * 256.0f); i1 = i1 > 255 ? 255 : (i1 < 0 ? 0 : i1); atomicAdd(&lh[i1], 1.0f); }
            if (m.z) { int i2 = (int)(v.z * 256.0f); i2 = i2 > 255 ? 255 : (i2 < 0 ? 0 : i2); atomicAdd(&lh[i2], 1.0f); }
            if (m.w) { int i3 = (int)(v.w * 256.0f); i3 = i3 > 255 ? 255 : (i3 < 0 ? 0 : i3); atomicAdd(&lh[i3], 1.0f); }
        }
    }
    __syncthreads();
    atomicAdd(&hist[b * NBINS + tid], lh[tid]);
}

// ---------------------------------------------------------------------------
// 3) normalized inclusive cumsum of hist and hist*arange via WMMA.
//    View hist as 16x16 matrix H; C = (Lstrict x H) x Ones + H x U is the
//    256-long inclusive prefix sum, done with chained V_WMMA_F32_16X16X4_F32.
//    One wave per batch; EXEC all ones as WMMA requires.
// ---------------------------------------------------------------------------
__global__ void wmma_cumsum_kernel(const float* __restrict__ hist,
                                   float* __restrict__ ch,
                                   float* __restrict__ cm) {
    __shared__ float lds[512];
    const int b    = blockIdx.x;
    const int lane = threadIdx.x;       // 0..31
    const int lo   = lane & 15;
    const int hi   = lane >> 4;         // 0 or 1
    const float* Hb = hist + b * NBINS;

    v2f hA[4], hB[4], wA[4], wB[4], lA[4], uB[4], onesB;
    onesB[0] = 1.0f; onesB[1] = 1.0f;
    #pragma unroll
    for (int c = 0; c < 4; ++c) {
        #pragma unroll
        for (int r = 0; r < 2; ++r) {
            const int k  = 4 * c + r + 2 * hi;
            const float av = Hb[lo * 16 + k];   // A element (M=lo, K=k)
            const float bv = Hb[k * 16 + lo];   // B element (K=k, N=lo)
            hA[c][r] = av;
            hB[c][r] = bv;
            wA[c][r] = av * (float)(lo * 16 + k);   // H .* flat-index
            wB[c][r] = bv * (float)(k * 16 + lo);
            lA[c][r] = (k <  lo) ? 1.0f : 0.0f;     // Lstrict[M][K] = (K < M)
            uB[c][r] = (k <= lo) ? 1.0f : 0.0f;     // U[K][N]      = (K <= N)
        }
    }

    float invTotal = 0.0f;
    #pragma unroll
    for (int pass = 0; pass < 2; ++pass) {
        const v2f* XA = pass ? wA : hA;
        const v2f* XB = pass ? wB : hB;

        v8f d1 = {};   // Lstrict x X
        v8f s  = {};   // X x U
        #pragma unroll
        for (int c = 0; c < 4; ++c) {
            d1 = __builtin_amdgcn_wmma_f32_16x16x4_f32(false, lA[c], false, XB[c],
                                                       (short)0, d1, false, false);
            s  = __builtin_amdgcn_wmma_f32_16x16x4_f32(false, XA[c], false, uB[c],
                                                       (short)0, s,  false, false);
        }
        #pragma unroll
        for (int r = 0; r < 8; ++r)
            lds[(r + 8 * hi) * 16 + lo] = d1[r];     // D-layout -> LDS
        __syncthreads();
        v8f csum = s;
        #pragma unroll
        for (int c = 0; c < 4; ++c) {
            const int k = 4 * c + 2 * hi;
            v2f d1a;
            d1a[0] = lds[lo * 16 + k + 0];           // reload in A-layout
            d1a[1] = lds[lo * 16 + k + 1];
            csum = __builtin_amdgcn_wmma_f32_16x16x4_f32(false, d1a, false, onesB,
                                                         (short)0, csum, false, false);
        }
        __syncthreads();
        #pragma unroll
        for (int r = 0; r < 8; ++r)
            lds[256 + (r + 8 * hi) * 16 + lo] = csum[r];
        __syncthreads();
        if (pass == 0) invTotal = 1.0f / lds[256 + 255];  // sum(hist); NaN if empty as in ref
        float* dst = (pass ? cm : ch) + b * NBINS;
        #pragma unroll
        for (int r = 0; r < 8; ++r)
            dst[(r + 8 * hi) * 16 + lo] = csum[r] * invTotal;
        __syncthreads();
    }
}

// ---------------------------------------------------------------------------
// 4) exhaustive 254x254 between-class-variance argmax per batch
//    (first-max tie-break to match jnp.argmax)
// ---------------------------------------------------------------------------
__global__ void argmax_kernel(const float* __restrict__ ch,
                              const float* __restrict__ cm,
                              float* __restrict__ thr) {
    __shared__ float sch[NBINS], scm[NBINS];
    __shared__ float sbv[256];
    __shared__ int   sidx[256];
    const int b   = blockIdx.x;
    const int tid = threadIdx.x;
    sch[tid] = ch[b * NBINS + tid];
    scm[tid] = cm[b * NBINS + tid];
    __syncthreads();
    const float tm  = scm[NBINS - 1];
    const float eps = 1e-8f;
    float bestv = -1.0f;
    int   bestp = 0;
    for (int p = tid; p < NT * NT; p += 256) {
        const int i = p / NT;
        const int j = p - i * NT;
        const float w0 = sch[i];
        const float cb = sch[j];
        const float w1 = cb - w0;
        const float w2 = 1.0f - cb;
        const float m0 = scm[i];
        const float m1 = scm[j];
        const float d0 = m0 / (w0 + eps) - tm;
        const float d1 = (m1 - m0) / (w1 + eps) - tm;
        const float d2 = (tm - m1) / (w2 + eps) - tm;
        float bv = w0 * d0 * d0 + w1 * d1 * d1 + w2 * d2 * d2;
        bv = (w0 > 0.0f && w1 > 0.0f && w2 > 0.0f) ? bv : 0.0f;
        if (bv > bestv) { bestv = bv; bestp = p; }   // p strictly increases
    }
    sbv[tid]  = bestv;
    sidx[tid] = bestp;
    __syncthreads();
    for (int s2 = 128; s2 > 0; s2 >>= 1) {
        if (tid < s2) {
            const float ov = sbv[tid + s2];
            const int   oi = sidx[tid + s2];
            if (ov > sbv[tid] || (ov == sbv[tid] && oi < sidx[tid])) {
                sbv[tid] = ov; sidx[tid] = oi;
            }
        }
        __syncthreads();
    }
    if (tid == 0) {
        const int am = sidx[0];
        thr[b * 2 + 0] = (float)(am / NT + 1) * (1.0f / 255.0f);  // t1
        thr[b * 2 + 1] = (float)(am % NT + 1) * (1.0f / 255.0f);  // t2
    }
}

// ---------------------------------------------------------------------------
// 5) fused loss pass (float4/uchar4 streaming):
//    sq = sum (ci - preds*mask)^2, sm = sum mask. Off-mask pixels contribute
//    0 (im0=0 < t1, cpred=0), so they are skipped.
// ---------------------------------------------------------------------------
#define LCHUNKS 64
#define LPP     (NPIX / LCHUNKS)     // 4096 pixels per block

__global__ void loss_kernel(const float* __restrict__ preds,
                            const float* __restrict__ images,
                            const unsigned char* __restrict__ mask,
                            const float* __restrict__ thr,
                            float* __restrict__ acc) {
    __shared__ float s_sq[256], s_sm[256];
    const int b   = blockIdx.y;
    const int tid = threadIdx.x;
    const float t1 = thr[b * 2 + 0];
    const float t2 = thr[b * 2 + 1];
    const int base = blockIdx.x * LPP;
    const float4* pr4  = (const float4*)(preds  + (size_t)b * NPIX + base);
    const float4* im4  = (const float4*)(images + (size_t)b * NPIX + base);
    const uchar4* mk4  = (const uchar4*)(mask   + (size_t)b * NPIX + base);
    float sq = 0.0f, sm = 0.0f;
    for (int t = tid; t < LPP / 4; t += 256) {
        __builtin_prefetch(&im4[t + 256], 0, 0);
        __builtin_prefetch(&pr4[t + 256], 0, 0);
        const uchar4 m = mk4[t];
        if (m.x | m.y | m.z | m.w) {
            const float4 v = im4[t];
            const float4 p = pr4[t];
            if (m.x) { float ci = (v.x >= t2) ? 1.0f : ((v.x >= t1) ? 0.5f : 0.0f); float d = ci - p.x; sq += d * d; sm += 1.0f; }
            if (m.y) { float ci = (v.y >= t2) ? 1.0f : ((v.y >= t1) ? 0.5f : 0.0f); float d = ci - p.y; sq += d * d; sm += 1.0f; }
            if (m.z) { float ci = (v.z >= t2) ? 1.0f : ((v.z >= t1) ? 0.5f : 0.0f); float d = ci - p.z; sq += d * d; sm += 1.0f; }
            if (m.w) { float ci = (v.w >= t2) ? 1.0f : ((v.w >= t1) ? 0.5f : 0.0f); float d = ci - p.w; sq += d * d; sm += 1.0f; }
        }
    }
    s_sq[tid] = sq;
    s_sm[tid] = sm;
    __syncthreads();
    for (int s2 = 128; s2 > 0; s2 >>= 1) {
        if (tid < s2) {
            s_sq[tid] += s_sq[tid + s2];
            s_sm[tid] += s_sm[tid + s2];
        }
        __syncthreads();
    }
    if (tid == 0) {
        atomicAdd(&acc[b * 2 + 0], s_sq[0]);
        atomicAdd(&acc[b * 2 + 1], s_sm[0]);
    }
}

// ---------------------------------------------------------------------------
// 6) finalize: masked mean over valid batches -> scalar
// ---------------------------------------------------------------------------
__global__ void final_kernel(const float* __restrict__ acc, float* __restrict__ out) {
    if (threadIdx.x == 0 && blockIdx.x == 0) {
        float s = 0.0f;
        int cnt = 0;
        for (int b = 0; b < BATCH; ++b) {
            const float sq = acc[b * 2 + 0];
            const float sm = acc[b * 2 + 1];
            if (sm > 0.0f) { s += sq / (sm + 1e-8f); ++cnt; }
        }
        out[0] = (cnt > 0) ? (s / (float)cnt) : 0.0f;
    }
}

// ---------------------------------------------------------------------------
extern "C" void kernel_launch(void* const* d_in, const int* in_sizes, int n_in,
                              void* d_out, int out_size, void* d_ws, size_t ws_size,
                              hipStream_t stream) {
    const float* preds  = (const float*)d_in[0];
    const float* labels = (const float*)d_in[1];
    const float* images = (const float*)d_in[2];
    float* out = (float*)d_out;

    char* ws = (char*)d_ws;
    unsigned char* mask = (unsigned char*)ws;            // 8 MB
    size_t off = ((size_t)BATCH * NPIX + 255) & ~(size_t)255;
    float* hist = (float*)(ws + off); off += (size_t)BATCH * NBINS * sizeof(float);
    float* ch   = (float*)(ws + off); off += (size_t)BATCH * NBINS * sizeof(float);
    float* cm   = (float*)(ws + off); off += (size_t)BATCH * NBINS * sizeof(float);
    float* thr  = (float*)(ws + off); off += (size_t)BATCH * 2 * sizeof(float);
    float* acc  = (float*)(ws + off); off += (size_t)BATCH * 2 * sizeof(float);

    init_kernel<<<(BATCH * NBINS + 255) / 256, 256, 0, stream>>>(hist, acc);
    dilate_kernel<<<dim3(WDIM / TW, HDIM / TH, BATCH), 256, 0, stream>>>(labels, mask);
    hist_kernel<<<dim3(HCHUNKS, BATCH), 256, 0, stream>>>(images, mask, hist);
    wmma_cumsum_kernel<<<BATCH, 32, 0, stream>>>(hist, ch, cm);
    argmax_kernel<<<BATCH, 256, 0, stream>>>(ch, cm, thr);
    loss_kernel<<<dim3(LCHUNKS, BATCH), 256, 0, stream>>>(preds, images, mask, thr, acc);
    final_kernel<<<1, 32, 0, stream>>>(acc, out);
}